// QuaternionSPT_50818053046292
// MI455X (gfx1250) — compile-verified
//
#include <hip/hip_runtime.h>

// QuaternionSPT on MI455X (gfx1250, wave32).
// Sequential BFS => single-wave kernel; per-step 32 Hamilton products are
// batched as two V_WMMA_F32_16X16X4_F32 ops (qn = R(qu) * conj(q_edge)).

#define DEG 32
#define MAXN 20000

typedef __attribute__((ext_vector_type(2))) float v2f;
typedef __attribute__((ext_vector_type(8))) float v8f;

__global__ __launch_bounds__(256) void qspt_init(float* __restrict__ out,
                                                 int* __restrict__ queue,
                                                 const int* __restrict__ startp,
                                                 int N) {
  const int i = blockIdx.x * blockDim.x + threadIdx.x;
  const int start = *startp;
  if (i == 0) out[0] = (float)start;            // tuple element 0: start_node
  if (i < N) {                                   // identity quats everywhere
    out[1 + 4 * i + 0] = 1.0f;
    out[1 + 4 * i + 1] = 0.0f;
    out[1 + 4 * i + 2] = 0.0f;
    out[1 + 4 * i + 3] = 0.0f;
    queue[i] = (i == 0) ? start : 0;             // queue0 = zeros.at[0].set(start)
  }
}

__global__ __launch_bounds__(32) void qspt_bfs(const float* __restrict__ edge_attr,
                                               const int* __restrict__ dst,
                                               float* __restrict__ out,
                                               int* __restrict__ queue,
                                               const int* __restrict__ startp,
                                               int N) {
  __shared__ unsigned char visited[MAXN];   // 20 KB of the 320 KB WGP LDS
  __shared__ int   vv_s[DEG];               // sorted neighbor ids
  __shared__ float aq[DEG][4];              // sorted conjugated edge quats
  __shared__ float qnbuf[DEG][4];           // WMMA results, row-major per edge

  if (N > MAXN) return;

  const int lane  = threadIdx.x;            // 0..31, one wave32
  const int start = *startp;
  float* nq = out + 1;

  for (int i = lane; i < N; i += DEG) visited[i] = 0;
  __syncthreads();
  if (lane == 0) visited[start] = 1;
  __syncthreads();

  int tail = 1;
  const int half = lane >> 4;               // 0: lanes 0-15, 1: lanes 16-31
  const int m    = lane & 15;

  for (int head = 0; head < N; ++head) {
    const bool active = head < tail;        // uniform across the wave
    const int  u = queue[head];

    // u's quaternion is final by BFS order (written when u was enqueued).
    const float bw = nq[4 * u + 0];
    const float bx = nq[4 * u + 1];
    const float by = nq[4 * u + 2];
    const float bz = nq[4 * u + 3];

    // Lane-per-edge: u's edges are contiguous [u*DEG .. u*DEG+31].
    const int    e = u * DEG + lane;
    const float4 q = *reinterpret_cast<const float4*>(edge_attr + 4 * (long)e);
    const int    v = dst[e];

    // Stable ascending rank by err = 2*degrees(acos(clip(|w|,0,1))).
    const float aw  = fminf(fabsf(q.x), 1.0f);
    const float key = 114.59155902616465f * acosf(aw);   // 2 * 180/pi
    int rank = 0;
#pragma unroll
    for (int j = 0; j < DEG; ++j) {
      const float kj = __shfl(key, j, DEG);
      rank += (kj < key || (kj == key && j < lane)) ? 1 : 0;
    }

    // Scatter conjugate (== inverse for unit quats) into sorted slots.
    aq[rank][0] =  q.x;
    aq[rank][1] = -q.y;
    aq[rank][2] = -q.z;
    aq[rank][3] = -q.w;
    vv_s[rank] = v;
    __syncthreads();

    // --- WMMA fragments -----------------------------------------------------
    // A (16x4 f32 layout): lane<16 -> row m, K={0,1}; lane>=16 -> row m, K={2,3}.
    v2f A1, A2, Bf;
    A1.x = aq[m][2 * half];        A1.y = aq[m][2 * half + 1];        // edges 0..15
    A2.x = aq[16 + m][2 * half];   A2.y = aq[16 + m][2 * half + 1];   // edges 16..31

    // B (4x16): B[k][n] = coefficient of a[k] in component n of a (x) qu.
    //   k=0: { bw,  bx,  by,  bz}   k=1: {-bx,  bw, -bz,  by}
    //   k=2: {-by,  bz,  bw, -bx}   k=3: {-bz, -by,  bx,  bw}
    // Layout mirrors A: v0 holds rows K=0 (lanes 0-15) / K=2 (lanes 16-31),
    // v1 holds rows K=1 / K=3; columns n>=4 are zero.
    float b0 = 0.0f, b1 = 0.0f;
    if (m < 4) {
      if (half == 0) {
        b0 = (m == 0) ?  bw : (m == 1) ?  bx : (m == 2) ?  by :  bz;
        b1 = (m == 0) ? -bx : (m == 1) ?  bw : (m == 2) ? -bz :  by;
      } else {
        b0 = (m == 0) ? -by : (m == 1) ?  bz : (m == 2) ?  bw : -bx;
        b1 = (m == 0) ? -bz : (m == 1) ? -by : (m == 2) ?  bx :  bw;
      }
    }
    Bf.x = b0; Bf.y = b1;

    v8f cz = {};
    // D = A*B: D[row][0..3] = conj(q_row) (x) qu.  EXEC is all-1s here.
    v8f d1 = __builtin_amdgcn_wmma_f32_16x16x4_f32(false, A1, false, Bf,
                                                   (short)0, cz, false, false);
    v8f d2 = __builtin_amdgcn_wmma_f32_16x16x4_f32(false, A2, false, Bf,
                                                   (short)0, cz, false, false);

    // Unpack D (C/D layout: VGPR r -> M = r + 8*half, N = m) via LDS.
    if (m < 4) {
#pragma unroll
      for (int r = 0; r < 8; ++r) {
        const int M = r + 8 * half;
        qnbuf[M][m]      = d1[r];   // edges 0..15
        qnbuf[16 + M][m] = d2[r];   // edges 16..31
      }
    }
    __syncthreads();

    const int   myv = vv_s[lane];
    const float qn0 = qnbuf[lane][0];
    const float qn1 = qnbuf[lane][1];
    const float qn2 = qnbuf[lane][2];
    const float qn3 = qnbuf[lane][3];

    // Exact sequential-scan semantics: a neighbor is taken iff not already
    // visited and no lower-ranked duplicate exists in this scan.
    const bool vis = visited[myv] != 0;
    bool dup = false;
#pragma unroll
    for (int j = 0; j < DEG; ++j) {
      const int vj = __shfl(myv, j, DEG);
      dup = dup || (j < lane && vj == myv);
    }
    const bool     take   = active && !vis && !dup;
    const unsigned mask   = (unsigned)__ballot(take);
    const int      offset = __popc(mask & ((1u << lane) - 1u));
    if (take) {
      queue[tail + offset] = myv;
      visited[myv] = 1;
      float* p = nq + 4 * (long)myv;
      p[0] = qn0; p[1] = qn1; p[2] = qn2; p[3] = qn3;
    }
    tail += __popc(mask);               // uniform: ballot is wave-uniform

    // Prefetch the next frontier node's edge data (global_prefetch_b8).
    if (head + 1 < tail) {
      const int un = queue[head + 1];
      __builtin_prefetch(edge_attr + 4 * (long)(un * DEG + lane), 0, 1);
      __builtin_prefetch(dst + (long)(un * DEG + lane), 0, 1);
    }

    __threadfence_block();              // nq/queue stores -> next-iter loads
    __syncthreads();                    // LDS slot reuse
  }
}

extern "C" void kernel_launch(void* const* d_in, const int* in_sizes, int n_in,
                              void* d_out, int out_size, void* d_ws, size_t ws_size,
                              hipStream_t stream) {
  const float* edge_attr  = (const float*)d_in[0];   // [E,4] f32
  const int*   edge_index = (const int*)d_in[1];     // [2,E] i32
  const int*   startp     = (const int*)d_in[2];     // scalar

  const int E = in_sizes[0] / 4;
  const int N = E / DEG;                 // fixed out-degree graph
  const int* dst = edge_index + E;       // row 1 of edge_index

  float* out   = (float*)d_out;          // [1 + 4N] f32
  int*   queue = (int*)d_ws;             // N ints of scratch

  qspt_init<<<(N + 255) / 256, 256, 0, stream>>>(out, queue, startp, N);
  qspt_bfs<<<1, DEG, 0, stream>>>(edge_attr, dst, out, queue, startp, N);
}